// MultiAttentionHead_25572235281061
// MI455X (gfx1250) — compile-verified
//
#include <hip/hip_runtime.h>
#include <hip/hip_bf16.h>

typedef __attribute__((ext_vector_type(2))) float v2f;
typedef __attribute__((ext_vector_type(8))) float v8f;
typedef __attribute__((ext_vector_type(4))) int i32x4;

#define B_SZ   128
#define T_SZ   256
#define C_SZ   384
#define H_SZ   6
#define D_SZ   64
#define N3_SZ  1152
#define M_SZ   (B_SZ * T_SZ)          // 32768
// scale = 1/sqrt(C) (reference quirk: full embed dim, not head dim)
#define ATTN_SCALE 0.05103103630798288f

// CDNA5 async global->LDS path (ASYNCcnt-tracked), with fallback.
#if __has_builtin(__builtin_amdgcn_global_load_async_to_lds_b128) && \
    __has_builtin(__builtin_amdgcn_s_wait_asynccnt)
#define HAVE_ASYNC_LDS 1
#else
#define HAVE_ASYNC_LDS 0
#endif

#if HAVE_ASYNC_LDS
typedef __attribute__((address_space(1))) i32x4* g128_t;
typedef __attribute__((address_space(3))) i32x4* l128_t;
__device__ __forceinline__ void async_copy16(const float* g, float* l) {
  __builtin_amdgcn_global_load_async_to_lds_b128((g128_t)g, (l128_t)l, 0, 0);
}
#endif

__device__ __forceinline__ v8f wmma4(v2f a, v2f b, v8f c) {
  // (neg_a, A, neg_b, B, c_mod, C, reuse_a, reuse_b)
  return __builtin_amdgcn_wmma_f32_16x16x4_f32(false, a, false, b, (short)0, c,
                                               false, false);
}

// ---------------------------------------------------------------------------
// Tiled GEMM: C(M x N) = A(M x 384) * W(384 x N).  64x64 block, 128 threads
// (4 waves), each wave owns a 32x32 accumulator (2x2 WMMA C-tiles).
// Double-buffered LDS tiles: one barrier per K-slab, next slab prefetched
// into registers while the current slab feeds the WMMA chain.
// MODE 0: scatter output columns into K/Q/V buffers laid out [b][h][t][d]
//         (qkv split order in the reference source is K, Q, V).
// MODE 1: plain row-major store to O0 with ld = 384.
// ---------------------------------------------------------------------------
template <int MODE>
__global__ __launch_bounds__(128) void gemm64(const float* __restrict__ A,
                                              const float* __restrict__ W,
                                              float* __restrict__ O0,
                                              float* __restrict__ O1,
                                              float* __restrict__ O2,
                                              int ldb) {
  __shared__ float As[2][64][20];  // stride 20: frag reads hit 16 banks
  __shared__ float Bs[2][16][64];

  const int bm = blockIdx.y * 64;
  const int bn = blockIdx.x * 64;
  const int tid = threadIdx.x;
  const int wave = tid >> 5;
  const int lane = tid & 31;
  const int lo = lane & 15;
  const int hi = lane >> 4;
  const int wm = (wave >> 1) * 32;
  const int wn = (wave & 1) * 32;

  const int ar = tid >> 2, ac = (tid & 3) << 2;   // A tile fill: 64x16
  const int br = tid >> 4, bc = (tid & 15) << 2;  // B tile fill: 16x64

  v8f acc[2][2] = {};

  // stage K-slab 0
#pragma unroll
  for (int p = 0; p < 2; ++p) {
    int row = ar + 32 * p;
    *(float4*)&As[0][row][ac] =
        *(const float4*)&A[(size_t)(bm + row) * 384 + ac];
  }
#pragma unroll
  for (int p = 0; p < 2; ++p) {
    int row = br + 8 * p;
    *(float4*)&Bs[0][row][bc] = *(const float4*)&W[(size_t)row * ldb + bn + bc];
  }
  __syncthreads();

  int cur = 0;
  for (int k0 = 16; k0 <= 384; k0 += 16) {
    float4 pa[2], pb[2];
    const bool more = (k0 < 384);
    if (more) {  // prefetch next slab into registers
#pragma unroll
      for (int p = 0; p < 2; ++p)
        pa[p] = *(const float4*)&A[(size_t)(bm + ar + 32 * p) * 384 + k0 + ac];
#pragma unroll
      for (int p = 0; p < 2; ++p)
        pb[p] = *(const float4*)&W[(size_t)(k0 + br + 8 * p) * ldb + bn + bc];
    }

#pragma unroll
    for (int s = 0; s < 4; ++s) {
      const int kk = 4 * s;
      v2f a0 = *(const v2f*)&As[cur][wm + lo][kk + 2 * hi];
      v2f a1 = *(const v2f*)&As[cur][wm + 16 + lo][kk + 2 * hi];
      v2f b0, b1;
      b0.x = Bs[cur][kk + 2 * hi][wn + lo];
      b0.y = Bs[cur][kk + 1 + 2 * hi][wn + lo];
      b1.x = Bs[cur][kk + 2 * hi][wn + 16 + lo];
      b1.y = Bs[cur][kk + 1 + 2 * hi][wn + 16 + lo];
      acc[0][0] = wmma4(a0, b0, acc[0][0]);
      acc[0][1] = wmma4(a0, b1, acc[0][1]);
      acc[1][0] = wmma4(a1, b0, acc[1][0]);
      acc[1][1] = wmma4(a1, b1, acc[1][1]);
    }

    if (more) {
      const int nxt = cur ^ 1;
#pragma unroll
      for (int p = 0; p < 2; ++p) *(float4*)&As[nxt][ar + 32 * p][ac] = pa[p];
#pragma unroll
      for (int p = 0; p < 2; ++p) *(float4*)&Bs[nxt][br + 8 * p][bc] = pb[p];
      __syncthreads();
      cur = nxt;
    }
  }

  if (MODE == 0) {
    // whole 64-wide N block lies inside one (section, head)
    const int sec = bn / 384;              // 0=K 1=Q 2=V
    const int h = (bn % 384) >> 6;         // head
    float* dst = (sec == 0) ? O0 : (sec == 1) ? O1 : O2;
#pragma unroll
    for (int i = 0; i < 2; ++i)
#pragma unroll
      for (int j = 0; j < 2; ++j)
#pragma unroll
        for (int e = 0; e < 8; ++e) {
          int gm = bm + wm + i * 16 + e + 8 * hi;  // global row (b*256+t)
          int dd = wn + j * 16 + lo;               // within-head col
          int bi = gm >> 8, tt = gm & 255;
          dst[(((size_t)bi * H_SZ + h) * T_SZ + tt) * D_SZ + dd] =
              acc[i][j][e];
        }
  } else {
    (void)O1; (void)O2;
#pragma unroll
    for (int i = 0; i < 2; ++i)
#pragma unroll
      for (int j = 0; j < 2; ++j)
#pragma unroll
        for (int e = 0; e < 8; ++e) {
          int gm = bm + wm + i * 16 + e + 8 * hi;
          int gn = bn + wn + j * 16 + lo;
          O0[(size_t)gm * 384 + gn] = acc[i][j][e];
        }
  }
}

// ---------------------------------------------------------------------------
// Attention: one block per (q-block of 64 rows, head, batch).  128 threads =
// 4 waves, each wave owns 16 query rows.  K (padded) + V + per-wave score
// rows live entirely in LDS (~197 KB of the WGP's 320 KB).  K/V staged with
// GLOBAL_LOAD_ASYNC_TO_LDS_B128 (ASYNCcnt) when available, overlapped with
// the Q-fragment register loads.
// ---------------------------------------------------------------------------
#define KS_STRIDE 68   // 64 + 4 pad: frag reads map to distinct banks
#define SS_STRIDE 260  // 256 + 4 pad
#define LDS_FLOATS (256 * KS_STRIDE + 256 * 64 + 4 * 16 * SS_STRIDE)

__global__ __launch_bounds__(128) void attn_kernel(
    const float* __restrict__ Kb, const float* __restrict__ Qb,
    const float* __restrict__ Vb, float* __restrict__ Ab) {
  extern __shared__ float smem[];
  float* Ks = smem;                  // [256][68]
  float* Vs = Ks + 256 * KS_STRIDE;  // [256][64]
  float* Sw = Vs + 256 * 64;         // [4][16][260]

  const int qblk = blockIdx.x;  // 0..3
  const int h = blockIdx.y;     // 0..5
  const int b = blockIdx.z;     // 0..127
  const int tid = threadIdx.x;
  const int wave = tid >> 5;
  const int lane = tid & 31;
  const int lo = lane & 15;
  const int hi = lane >> 4;

  const size_t head_base = ((size_t)b * H_SZ + h) * T_SZ * D_SZ;
  const float* Kg = Kb + head_base;
  const float* Vg = Vb + head_base;
  const float* Qg = Qb + head_base + (size_t)(qblk * 64) * D_SZ;

  // Stage K (padded stride 68) and V into LDS.
#if HAVE_ASYNC_LDS
  for (int idx = tid; idx < 256 * 16; idx += 128) {
    int r = idx >> 4, c4 = (idx & 15) << 2;
    async_copy16(&Kg[r * 64 + c4], &Ks[r * KS_STRIDE + c4]);
    async_copy16(&Vg[r * 64 + c4], &Vs[r * 64 + c4]);
  }
#else
  for (int idx = tid; idx < 256 * 16; idx += 128) {
    int r = idx >> 4, c4 = (idx & 15) << 2;
    *(float4*)&Ks[r * KS_STRIDE + c4] = *(const float4*)&Kg[r * 64 + c4];
    *(float4*)&Vs[r * 64 + c4] = *(const float4*)&Vg[r * 64 + c4];
  }
#endif

  // Q A-fragments for all 16 k-steps (d = 64) held in registers; these
  // global loads overlap the async K/V staging.
  v2f q[16];
#pragma unroll
  for (int s = 0; s < 16; ++s)
    q[s] = *(const v2f*)&Qg[(wave * 16 + lo) * 64 + 4 * s + 2 * hi];

#if HAVE_ASYNC_LDS
  __builtin_amdgcn_s_wait_asynccnt(0);
#endif
  __syncthreads();

  float* S = Sw + wave * 16 * SS_STRIDE;
  const int qt0 = qblk * 64 + wave * 16;  // first global q row of this wave

  // Phase A: S = scale * Q K^T with causal mask, written to LDS.
  for (int n0 = 0; n0 < 256; n0 += 16) {
    v8f acc = {};
#pragma unroll
    for (int s = 0; s < 16; ++s) {
      const int kk = 4 * s;
      v2f bb = *(const v2f*)&Ks[(n0 + lo) * KS_STRIDE + kk + 2 * hi];
      acc = wmma4(q[s], bb, acc);
    }
    const int kt = n0 + lo;
#pragma unroll
    for (int e = 0; e < 8; ++e) {
      int r = e + 8 * hi;
      float v = acc[e] * ATTN_SCALE;
      if (kt > qt0 + r) v = -__builtin_inff();
      S[r * SS_STRIDE + n0 + lo] = v;
    }
  }

  // Phase B: row softmax in LDS; 2 lanes per row (wave32, same wave => LDS
  // ops stay in order, no barrier needed).
  {
    const int r = lane >> 1, half = lane & 1;
    float* row = S + r * SS_STRIDE + half * 128;
    float m = -__builtin_inff();
    for (int i = 0; i < 128; ++i) m = fmaxf(m, row[i]);
    m = fmaxf(m, __shfl_xor(m, 1, 32));
    float sum = 0.0f;
    for (int i = 0; i < 128; ++i) {
      float e = __expf(row[i] - m);
      row[i] = e;
      sum += e;
    }
    sum += __shfl_xor(sum, 1, 32);
    const float inv = 1.0f / sum;
    for (int i = 0; i < 128; ++i) row[i] *= inv;
  }

  // Phase C: O = P V  (K-dim 256), P read back from LDS in A-frag layout.
  v8f acc[4] = {};
#pragma unroll 4
  for (int s = 0; s < 64; ++s) {
    const int kk = 4 * s;
    v2f a = *(const v2f*)&S[lo * SS_STRIDE + kk + 2 * hi];
#pragma unroll
    for (int j = 0; j < 4; ++j) {
      v2f bb;
      bb.x = Vs[(kk + 2 * hi) * 64 + 16 * j + lo];
      bb.y = Vs[(kk + 1 + 2 * hi) * 64 + 16 * j + lo];
      acc[j] = wmma4(a, bb, acc[j]);
    }
  }

  // Write attention output in [b][t][c] layout (c = h*64 + dd).
#pragma unroll
  for (int j = 0; j < 4; ++j)
#pragma unroll
    for (int e = 0; e < 8; ++e) {
      int t = qt0 + e + 8 * hi;
      Ab[((size_t)b * T_SZ + t) * C_SZ + h * 64 + 16 * j + lo] = acc[j][e];
    }
}

// ---------------------------------------------------------------------------
extern "C" void kernel_launch(void* const* d_in, const int* in_sizes, int n_in,
                              void* d_out, int out_size, void* d_ws,
                              size_t ws_size, hipStream_t stream) {
  const float* X = (const float*)d_in[0];      // (128, 256, 384)
  const float* Wqkv = (const float*)d_in[1];   // (384, 1152)
  const float* Wproj = (const float*)d_in[2];  // (384, 384)
  float* out = (float*)d_out;                  // (128, 256, 384)

  const size_t BHTD = (size_t)B_SZ * H_SZ * T_SZ * D_SZ;  // 12.58M floats
  float* Kbuf = (float*)d_ws;
  float* Qbuf = Kbuf + BHTD;
  float* Vbuf = Qbuf + BHTD;
  float* Abuf = Vbuf + BHTD;  // [b][t][c]

  // 1) fused QKV projection, scattered into [b][h][t][d] K/Q/V buffers
  dim3 g1(N3_SZ / 64, M_SZ / 64);
  gemm64<0><<<g1, 128, 0, stream>>>(X, Wqkv, Kbuf, Qbuf, Vbuf, N3_SZ);

  // 2) causal attention per (q-block, head, batch)
  dim3 g2(T_SZ / 64, H_SZ, B_SZ);
  attn_kernel<<<g2, 128, LDS_FLOATS * sizeof(float), stream>>>(Kbuf, Qbuf,
                                                               Vbuf, Abuf);

  // 3) output projection
  dim3 g3(C_SZ / 64, M_SZ / 64);
  gemm64<1><<<g3, 128, 0, stream>>>(Abuf, Wproj, out, nullptr, nullptr, C_SZ);
}